// maxpool_approx_module_5970004541569
// MI455X (gfx1250) — compile-verified
//
#include <hip/hip_runtime.h>

// Problem geometry (from reference):
//   x: (32, 64, 112, 112) fp32 -> out: (32, 64, 56, 56) fp32
//   Flattened: 114688 output rows of 56; output row r consumes input
//   floats [224*r, 224*r + 224)  (two full 112-wide input rows).
#define OUT_ROWS      114688
#define TILE_ROWS     32                       // output rows per tile
#define NTILES        (OUT_ROWS / TILE_ROWS)   // 3584
#define GRID          896                      // persistent blocks (4 tiles each)
#define BLOCK         256                      // 8 waves (wave32)
#define TILE_IN       (TILE_ROWS * 224)        // 7168 floats  = 28672 B
#define TILE_CHUNKS   (TILE_IN / 4)            // 1792 x B128 chunks
#define CHUNK_ITERS   (TILE_CHUNKS / BLOCK)    // 7  (uniform across all waves)
#define TILE_OUT      (TILE_ROWS * 56)         // 1792 outputs
#define OUT_ITERS     (TILE_OUT / BLOCK)       // 7

// ---- CDNA5 async global->LDS path (guarded probes) ----
#if defined(__has_builtin)
#  if __has_builtin(__builtin_amdgcn_global_load_async_to_lds_b128)
#    define USE_ASYNC 1
#  endif
#endif
#ifndef USE_ASYNC
#  define USE_ASYNC 0
#endif

// Types matching the builtin's parameters (per hipcc diagnostic:
// param0 is "__attribute__((__vector_size__(4*sizeof(int)))) int __device__ *",
// i.e. GCC-vector int4 in addrspace(1)).
typedef int v4i __attribute__((vector_size(16)));
typedef v4i __attribute__((address_space(1))) v4i_g;   // global
typedef v4i __attribute__((address_space(3))) v4i_l;   // LDS

template <int N>
__device__ __forceinline__ void wait_async_le() {
#if USE_ASYNC
#  if defined(__has_builtin) && __has_builtin(__builtin_amdgcn_s_wait_asynccnt)
  __builtin_amdgcn_s_wait_asynccnt(N);
#  else
  asm volatile("s_wait_asynccnt %0" :: "n"(N) : "memory");
#  endif
#endif
}

__device__ __forceinline__ void issue_tile(float* __restrict__ dst,
                                           const float* __restrict__ x,
                                           int tile, int tid) {
  const float* src = x + (size_t)tile * TILE_IN;   // 16B-aligned
#pragma unroll
  for (int i = 0; i < CHUNK_ITERS; ++i) {
    const int q = tid + i * BLOCK;                 // 0..1791, all lanes active
#if USE_ASYNC
    __builtin_amdgcn_global_load_async_to_lds_b128(
        (v4i_g*)(src + 4 * q),
        (v4i_l*)(dst + 4 * q),
        0, 0);
#else
    *(float4*)(dst + 4 * q) = *(const float4*)(src + 4 * q);
#endif
  }
}

__device__ __forceinline__ float poly8(float v, const float c[8]) {
  float acc = c[7];
#pragma unroll
  for (int k = 6; k >= 0; --k) acc = fmaf(acc, v, c[k]);
  return acc;
}

__device__ __forceinline__ float max_approx(float a, float b,
                                            const float c0[8], const float c1[8]) {
  const float d = a - b;
  const float s = poly8(poly8(d, c0), c1);
  return 0.5f * (a + b + d * s);
}

__global__ __launch_bounds__(BLOCK)
void maxpool_poly_kernel(const float* __restrict__ x,
                         const float* __restrict__ c0g,
                         const float* __restrict__ c1g,
                         float* __restrict__ out) {
  __shared__ float buf[2][TILE_IN];                // 2 x 28672 B, double-buffered
  const int tid = threadIdx.x;

  // Coefficients: uniform addresses -> scalar loads, live in registers.
  float c0[8], c1[8];
#pragma unroll
  for (int k = 0; k < 8; ++k) { c0[k] = c0g[k]; c1[k] = c1g[k]; }

  constexpr float SF  = 0.499755859375f;                  // (1 - 4*0.5^13)/2
  constexpr float INV = (float)(1.0 / 0.499755859375);    // 1/SF

  int cur = 0;
  issue_tile(buf[0], x, (int)blockIdx.x, tid);            // prologue DMA

  for (int t = blockIdx.x; t < NTILES; t += GRID) {
    const int tn = t + GRID;
    if (tn < NTILES) {
      issue_tile(buf[cur ^ 1], x, tn, tid);               // overlap next tile's DMA
      wait_async_le<CHUNK_ITERS>();                       // current tile complete (in-order)
    } else {
      wait_async_le<0>();
    }
    __syncthreads();                                      // LDS visible to all waves

    const float* b = buf[cur];
    float* orow = out + (size_t)t * TILE_OUT;
#pragma unroll
    for (int i = 0; i < OUT_ITERS; ++i) {
      const int o  = tid + i * BLOCK;                     // 0..1791
      const int lr = o / 56;                              // local output row
      const int j  = o - lr * 56;                         // output column
      // window: input rows 2*lr, 2*lr+1 at cols 2j, 2j+1 (8B-aligned float2, no bank conflicts)
      const float2 lo = *(const float2*)&b[lr * 224 + 2 * j];
      const float2 hi = *(const float2*)&b[lr * 224 + 112 + 2 * j];
      const float e0 = fmaf(lo.x, SF, 0.5f);
      const float e1 = fmaf(lo.y, SF, 0.5f);
      const float e2 = fmaf(hi.x, SF, 0.5f);
      const float e3 = fmaf(hi.y, SF, 0.5f);
      const float s1 = max_approx(e0, e1, c0, c1);
      const float s2 = max_approx(e2, e3, c0, c1);
      const float r  = max_approx(s1, s2, c0, c1);
      orow[o] = (r - 0.5f) * INV;                         // coalesced fp32 store
    }
    __syncthreads();                                      // done reading before next DMA lands
    cur ^= 1;
  }
}

extern "C" void kernel_launch(void* const* d_in, const int* in_sizes, int n_in,
                              void* d_out, int out_size, void* d_ws, size_t ws_size,
                              hipStream_t stream) {
  (void)in_sizes; (void)n_in; (void)d_ws; (void)ws_size; (void)out_size;
  const float* x  = (const float*)d_in[0];
  const float* c0 = (const float*)d_in[1];
  const float* c1 = (const float*)d_in[2];
  float* out      = (float*)d_out;
  maxpool_poly_kernel<<<dim3(GRID), dim3(BLOCK), 0, stream>>>(x, c0, c1, out);
}